// NeRVFrontToBackInverseRenderer_40681930227897
// MI455X (gfx1250) — compile-verified
//
#include <hip/hip_runtime.h>
#include <stdint.h>

#define VDIM 256
#define NVOX (VDIM * VDIM * VDIM)   // 16,777,216 voxels per output volume
#define BLOCK 256
#define CHUNKS_PER_BLOCK 8
#define NBUF 3                      // triple buffer: 1 barrier/iter is then safe

// ---------------------------------------------------------------------------
// CDNA5 async global->LDS copy (ASYNCcnt path). The builtin is declared on
// this toolchain with (AS1 int*, AS3 int*, imm offset, imm cpol).
// ---------------------------------------------------------------------------
#if defined(__has_builtin)
#  if __has_builtin(__builtin_amdgcn_global_load_async_to_lds_b32) && \
      __has_builtin(__builtin_amdgcn_s_wait_asynccnt)
#    define USE_ASYNC_BUILTIN 1
#  endif
#endif

typedef __attribute__((address_space(1))) int as1_int;
typedef __attribute__((address_space(3))) int as3_int;

__device__ __forceinline__ void async_cp_b32(const float* gsrc, float* ldst) {
#ifdef USE_ASYNC_BUILTIN
  __builtin_amdgcn_global_load_async_to_lds_b32(
      (as1_int*)gsrc, (as3_int*)ldst, 0, 0);
#else
  unsigned loff = (unsigned)(size_t)(as3_int*)ldst;
  asm volatile("global_load_async_to_lds_b32 %0, %1, off"
               :: "v"(loff), "v"(gsrc) : "memory");
#endif
}

__device__ __forceinline__ void wait_async_le4() {
#ifdef USE_ASYNC_BUILTIN
  __builtin_amdgcn_s_wait_asynccnt(4);
#else
  asm volatile("s_wait_asynccnt 0x4" ::: "memory");
#endif
}

__device__ __forceinline__ void wait_async_0() {
#ifdef USE_ASYNC_BUILTIN
  __builtin_amdgcn_s_wait_asynccnt(0);
#else
  asm volatile("s_wait_asynccnt 0x0" ::: "memory");
#endif
}

// Non-returning agent-scope f32 atomic add -> global_atomic_add_f32 (L2 RMW).
__device__ __forceinline__ void atomic_add_f32(float* p, float v) {
  __hip_atomic_fetch_add(p, v, __ATOMIC_RELAXED, __HIP_MEMORY_SCOPE_AGENT);
}

// ---------------------------------------------------------------------------
// Zero the 2 output volumes with 128-bit stores (harness poisons d_out).
// ---------------------------------------------------------------------------
__global__ void __launch_bounds__(BLOCK)
zero_kernel(float4* __restrict__ out, int n4) {
  int i = blockIdx.x * BLOCK + threadIdx.x;
  if (i < n4) out[i] = make_float4(0.f, 0.f, 0.f, 0.f);
}

// ---------------------------------------------------------------------------
// Trilinear splat. Each block consumes CHUNKS_PER_BLOCK chunks of 256 points.
// Point xyz+feature are staged into LDS via async copies (triple-buffered),
// then every lane does 8 global f32 atomics into d_out. The view-mean is
// folded in: batches 0,1 accumulate 0.5*w*f into volume 0; batch 2
// accumulates w*f into volume 1.
// ---------------------------------------------------------------------------
__global__ void __launch_bounds__(BLOCK)
splat_kernel(const float* __restrict__ pts,   // [B*P*3]
             const float* __restrict__ feat,  // [B*P]
             float* __restrict__ out,         // [2*NVOX]
             int P, int NP) {
  __shared__ float sPts[NBUF][3 * BLOCK];
  __shared__ float sFeat[NBUF][BLOCK];

  const int t = threadIdx.x;
  const int nChunksTotal = NP / BLOCK;  // NP is a multiple of 256 here
  const int chunk0 = blockIdx.x * CHUNKS_PER_BLOCK;
  int myChunks = nChunksTotal - chunk0;
  if (myChunks <= 0) return;
  if (myChunks > CHUNKS_PER_BLOCK) myChunks = CHUNKS_PER_BLOCK;

  auto prefetch = [&](int it) {
    const int c = chunk0 + it;
    const float* pbase = pts + (size_t)c * (3 * BLOCK);
    const float* fbase = feat + (size_t)c * BLOCK;
    float* dp = sPts[it % NBUF];
    // 768 dwords of xyz, dword-interleaved across the 256 threads (3 each),
    // plus 1 feature dword per thread: 4 ASYNCcnt ops per wave per chunk.
    async_cp_b32(pbase + t,       dp + t);
    async_cp_b32(pbase + t + 256, dp + t + 256);
    async_cp_b32(pbase + t + 512, dp + t + 512);
    async_cp_b32(fbase + t, sFeat[it % NBUF] + t);
  };

  prefetch(0);
  for (int it = 0; it < myChunks; ++it) {
    if (it + 1 < myChunks) {
      prefetch(it + 1);   // writes buf (it+1)%3, consumed last at it-2: safe
      wait_async_le4();   // in-order: <=4 outstanding => chunk `it` landed
    } else {
      wait_async_0();
    }
    __syncthreads();      // other waves' copies for chunk `it` also landed

    const float* cp = sPts[it % NBUF];
    const float px = cp[3 * t + 0];
    const float py = cp[3 * t + 1];
    const float pz = cp[3 * t + 2];
    const float f  = sFeat[it % NBUF][t];

    const int pointIdx = (chunk0 + it) * BLOCK + t;
    const bool firstGroup = pointIdx < 2 * P;        // views {0,1} -> mean
    const float fs = firstGroup ? 0.5f * f : f;
    const unsigned base = firstGroup ? 0u : (unsigned)NVOX;

    // align_corners=True mapping: v = (p + 1) * (V-1)/2
    const float x = fmaf(px, 127.5f, 127.5f);
    const float y = fmaf(py, 127.5f, 127.5f);
    const float z = fmaf(pz, 127.5f, 127.5f);
    const float flx = floorf(x), fly = floorf(y), flz = floorf(z);
    const int ix = (int)flx, iy = (int)fly, iz = (int)flz;
    const float fx = x - flx, fy = y - fly, fz = z - flz;
    const float wx0 = 1.0f - fx;

#pragma unroll
    for (int dz = 0; dz < 2; ++dz) {
      const int Z = iz + dz;
      if ((unsigned)Z >= (unsigned)VDIM) continue;
      const float az = fs * (dz ? fz : (1.0f - fz));
#pragma unroll
      for (int dy = 0; dy < 2; ++dy) {
        const int Y = iy + dy;
        if ((unsigned)Y >= (unsigned)VDIM) continue;
        const float ayz = az * (dy ? fy : (1.0f - fy));
        const unsigned rowBase =
            base + ((unsigned)Z * VDIM + (unsigned)Y) * VDIM;
        const int X0 = ix;
        if ((unsigned)X0 < (unsigned)VDIM)
          atomic_add_f32(out + rowBase + (unsigned)X0, ayz * wx0);
        const int X1 = ix + 1;
        if ((unsigned)X1 < (unsigned)VDIM)
          atomic_add_f32(out + rowBase + (unsigned)X1, ayz * fx);
      }
    }
    // No trailing barrier needed: next prefetch targets buf (it+2)%3, whose
    // last consumer finished before the barrier above (triple buffering).
  }
}

// ---------------------------------------------------------------------------
// d_in[0]: points_3d  [B,P,3] f32   d_in[1]: points_features [B,P,1] f32
// d_out  : [2,1,256,256,256] f32
// ---------------------------------------------------------------------------
extern "C" void kernel_launch(void* const* d_in, const int* in_sizes, int n_in,
                              void* d_out, int out_size, void* d_ws,
                              size_t ws_size, hipStream_t stream) {
  const float* pts  = (const float*)d_in[0];
  const float* feat = (const float*)d_in[1];
  float* out = (float*)d_out;

  const int NP = in_sizes[1];      // B*P point count (features are 1-channel)
  const int P  = NP / 3;           // B == sum(N_VIEWS) == 3

  const int n4 = out_size / 4;     // out_size = 2 * 256^3, divisible by 4
  zero_kernel<<<(n4 + BLOCK - 1) / BLOCK, BLOCK, 0, stream>>>((float4*)out, n4);

  const int nChunks = NP / BLOCK;
  const int nBlocks = (nChunks + CHUNKS_PER_BLOCK - 1) / CHUNKS_PER_BLOCK;
  splat_kernel<<<nBlocks, BLOCK, 0, stream>>>(pts, feat, out, P, NP);
}